// ClusteringAffinity_38671885533424
// MI455X (gfx1250) — compile-verified
//
#include <hip/hip_runtime.h>

// CDNA5 / gfx1250, wave32. D = A(16x4 f32) x B(4x16 f32) + C(16x16 f32).
typedef __attribute__((ext_vector_type(2))) float v2f;  // A/B: 64 elems / 32 lanes = 2 VGPRs
typedef __attribute__((ext_vector_type(8))) float v8f;  // C/D: 256 elems / 32 lanes = 8 VGPRs

#define SIGMA_F        10.0f
#define LOG2E_F        1.4426950408889634f
#define C_CLUSTERS     100
#define M_CENTERS      5
#define MC             500      // C * m
#define NTILES         50       // 100 classes, 2 per 16-row WMMA tile (m padded 5->8)
#define ROWS_PER_WAVE  16
#define WAVES_PER_BLK  8
#define ROWS_PER_BLK   (ROWS_PER_WAVE * WAVES_PER_BLK)   // 128
#define OSTRIDE        104      // padded row stride in LDS out-staging (dwords)
#define OUT_COLS       101

// ---------------------------------------------------------------------------
// Kernel 1: pairwise-center regularizer rw (500 centers, h=1) -> d_ws[0]
// ---------------------------------------------------------------------------
__global__ __launch_bounds__(256)
void rw_kernel(const float* __restrict__ W, float* __restrict__ rw_out) {
    __shared__ float w[MC];
    __shared__ float red[256];
    __shared__ float s_mu;
    const int tid = threadIdx.x;

    for (int i = tid; i < MC; i += 256) w[i] = W[i];
    __syncthreads();

    // pass 1: mu = denom * sum_{i<j} (w_i - w_j)^2
    float sum = 0.f;
    for (int i = tid; i < MC; i += 256) {
        const float wi = w[i];
        for (int j = i + 1; j < MC; ++j) {
            const float d = wi - w[j];
            sum += d * d;
        }
    }
    red[tid] = sum;
    __syncthreads();
    for (int s = 128; s > 0; s >>= 1) {
        if (tid < s) red[tid] += red[tid + s];
        __syncthreads();
    }
    const float denom = 2.0f / ((float)MC * (float)MC - (float)MC);
    if (tid == 0) s_mu = red[0] * denom;
    __syncthreads();
    const float mu = s_mu;

    // pass 2: rw = denom * sum_{i<j} ((w_i - w_j)^2 - mu)^2
    float sum2 = 0.f;
    for (int i = tid; i < MC; i += 256) {
        const float wi = w[i];
        for (int j = i + 1; j < MC; ++j) {
            const float d = wi - w[j];
            const float e = d * d - mu;
            sum2 += e * e;
        }
    }
    red[tid] = sum2;
    __syncthreads();
    for (int s = 128; s > 0; s >>= 1) {
        if (tid < s) red[tid] += red[tid + s];
        __syncthreads();
    }
    if (tid == 0) rw_out[0] = red[0] * denom;
}

// ---------------------------------------------------------------------------
// Kernel 2: distances via WMMA.
//   q = w^2 - 2*f*w  computed as one WMMA: D = A(w) x B(-2f) + C(w^2)
//   dist(n,c) = exp(-(f^2 + min_m q)/sigma) = max_m exp(-(f-w)^2/sigma)
// Tile layout (per WMMA, M = padded center index, N = sample index):
//   M = r + 8*hi  (r = D-VGPR index 0..7, hi = lane>=16)
//   class c = 2*t + hi, intra-class m = r  (m in [5,8) padded with w=0, w^2=1e30)
//   N = lane & 15 -> sample n_base + (lane&15)
// ---------------------------------------------------------------------------
__global__ __launch_bounds__(256)
void affinity_kernel(const float* __restrict__ f, const float* __restrict__ W,
                     const float* __restrict__ rw_ptr, float* __restrict__ out) {
    __shared__ __align__(16) float wpad  [NTILES * 16];   // A operand values
    __shared__ __align__(16) float wsqpad[NTILES * 16];   // C operand values
    __shared__ __align__(16) float ostage[WAVES_PER_BLK * ROWS_PER_WAVE * OSTRIDE];

    const int tid = threadIdx.x;

    // Stage W into WMMA-ready padded layout (m: 5 -> 8).
    for (int i = tid; i < NTILES * 16; i += 256) {
        const int t = i >> 4;
        const int M = i & 15;
        const int c = 2 * t + (M >> 3);
        const int m = M & 7;
        if (m < M_CENTERS) {
            const float w = W[c * M_CENTERS + m];
            wpad[i]   = w;
            wsqpad[i] = w * w;
        } else {
            wpad[i]   = 0.0f;    // contributes 0 to A x B
            wsqpad[i] = 1e30f;   // never wins the min
        }
    }
    const float rw = rw_ptr[0];
    __syncthreads();

    const int wave = tid >> 5;
    const int lane = tid & 31;
    const int hi   = lane >> 4;       // which 16-lane half
    const int ln   = lane & 15;       // sample index within tile

    const long n_base = (long)blockIdx.x * ROWS_PER_BLK + (long)wave * ROWS_PER_WAVE;
    const float fl  = f[n_base + ln];         // lanes 16-31 duplicate lanes 0-15
    const float fsq = fl * fl;

    // exp(-(fsq+mn)/sigma) = exp2((fsq+mn) * scl), scl = -log2e/sigma
    const float scl     = -(LOG2E_F / SIGMA_F);
    const float fsq_scl = fsq * scl;          // loop-invariant part

    // A-operand lane mask: zero out lanes 16-31 (they'd be K=2, must be 0).
    const float amask = hi ? 0.0f : 1.0f;

    // B matrix (4x16): row K=0 = -2*f in lanes 0-15 of first VGPR, rest zero.
    v2f B;
    B.x = (-2.0f * fl) * amask;
    B.y = 0.0f;

    float* ost = &ostage[wave * ROWS_PER_WAVE * OSTRIDE];

    for (int t = 0; t < NTILES; ++t) {
        // Unconditional LDS broadcast read, then mask -> no EXEC divergence.
        const float wv = wpad[t * 16 + ln];
        v2f A;
        A.x = wv * amask;
        A.y = 0.0f;

        // C matrix: C[r] (this lane) = w^2 at M = r + 8*hi  (two ds_load_b128).
        const float4* cp = (const float4*)&wsqpad[t * 16 + 8 * hi];
        const float4 c0 = cp[0];
        const float4 c1 = cp[1];
        v8f C;
        C[0] = c0.x; C[1] = c0.y; C[2] = c0.z; C[3] = c0.w;
        C[4] = c1.x; C[5] = c1.y; C[6] = c1.z; C[7] = c1.w;

        // D[r] = w_pad[M] * (-2 f_N) + w_pad[M]^2   (EXEC all ones, uniform flow)
        v8f D = __builtin_amdgcn_wmma_f32_16x16x4_f32(
            /*neg_a=*/false, A, /*neg_b=*/false, B,
            /*c_mod=*/(short)0, C, /*reuse_a=*/false, /*reuse_b=*/false);

        // min over the 8 padded m-slots (pure per-lane VGPR reduction)
        const float mn = fminf(fminf(fminf(D[0], D[1]), fminf(D[2], D[3])),
                               fminf(fminf(D[4], D[5]), fminf(D[6], D[7])));

        // one v_fma + one v_exp
        const float val = __builtin_amdgcn_exp2f(fmaf(mn, scl, fsq_scl));

        // stage transposed: row = sample, col = class (c = 2t + hi)
        ost[ln * OSTRIDE + 2 * t + hi] = val;
    }
    __syncthreads();

    // Coalesced write-out: each wave emits its 16 rows of 101 floats.
    for (int r = 0; r < ROWS_PER_WAVE; ++r) {
        float* __restrict__ orow = out + (n_base + r) * (long)OUT_COLS;
        const float* srow = &ost[r * OSTRIDE];
#pragma unroll
        for (int chunk = 0; chunk < 4; ++chunk) {
            const int col = chunk * 32 + lane;
            if (col < C_CLUSTERS)       orow[col] = srow[col];
            else if (col == C_CLUSTERS) orow[col] = rw;   // regularizer column
        }
    }
}

// ---------------------------------------------------------------------------
extern "C" void kernel_launch(void* const* d_in, const int* in_sizes, int n_in,
                              void* d_out, int out_size, void* d_ws, size_t ws_size,
                              hipStream_t stream) {
    const float* f = (const float*)d_in[0];   // (N, 1) float32, N = 262144
    const float* W = (const float*)d_in[1];   // (C, m, 1) = 500 float32
    float* out = (float*)d_out;               // (N, 101) float32
    float* rw  = (float*)d_ws;                // scratch scalar

    rw_kernel<<<1, 256, 0, stream>>>(W, rw);

    const int n = in_sizes[0];                          // 262144
    const int nblocks = (n + ROWS_PER_BLK - 1) / ROWS_PER_BLK;  // 2048
    affinity_kernel<<<nblocks, 256, 0, stream>>>(f, W, rw, out);
}